// EnergyIrrepModulation_32169305047130
// MI455X (gfx1250) — compile-verified
//
#include <hip/hip_runtime.h>

typedef __attribute__((ext_vector_type(2))) float v2f;
typedef __attribute__((ext_vector_type(8))) float v8f;
typedef __attribute__((ext_vector_type(4))) float f32x4;

// -------------------------------------------------------------------------
// Wave-per-tile fp32 GEMM using V_WMMA_F32_16X16X4_F32.
// out(Mpad x N) = act(A(M x K) @ W(K x N) + bias), Mpad = tiles_m*16.
// A-fragment (16x4 f32): lanes 0-15 hold M=lane,    VGPR0/1 = K={0,1};
//                        lanes 16-31 hold M=lane-16, VGPR0/1 = K={2,3}.
// B-fragment (4x16): row-striped across lanes (same half split).
// C/D (16x16 f32): VGPR j -> row j + 8*(lane>>4), col = lane&15.
// Pad-row handling: clamp the row address (always in-bounds) and scale the
// loaded A value by a 0/1 mask -> unconditional loads, no EXEC masking.
// -------------------------------------------------------------------------
template<bool RELU>
__global__ void gemm_wmma_f32(const float* __restrict__ A,
                              const float* __restrict__ W,
                              const float* __restrict__ bias,
                              float* __restrict__ out,
                              int M_actual, int K, int N, int tiles_n)
{
    const int tile = blockIdx.x;
    const int tm   = tile / tiles_n;
    const int tn   = tile % tiles_n;
    const int lane = threadIdx.x;      // one wave32 per block
    const int half = lane >> 4;        // 0 or 1
    const int l16  = lane & 15;
    const int row  = tm * 16 + l16;    // A row for this lane
    const int col  = tn * 16 + l16;    // B/C column for this lane

    const float amask   = (row < M_actual) ? 1.0f : 0.0f;     // one cndmask
    const int   rowsafe = (row < M_actual) ? row : 0;         // clamped addr
    const float* Arow = A + (size_t)rowsafe * K;

    v8f c = {};
    for (int k0 = 0; k0 < K; k0 += 4) {
        const int ka = k0 + 2 * half;
        // 8B-aligned pair load (ka even, K even)
        v2f a = *(const v2f*)(Arow + ka);
        a.x *= amask;
        a.y *= amask;
        v2f b;
        b.x = W[(ka)     * N + col];
        b.y = W[(ka + 1) * N + col];
        c = __builtin_amdgcn_wmma_f32_16x16x4_f32(
                /*neg_a=*/false, a, /*neg_b=*/false, b,
                /*c_mod=*/(short)0, c, /*reuse_a=*/false, /*reuse_b=*/false);
    }

    const float bv = bias[col];
    #pragma unroll
    for (int j = 0; j < 8; ++j) {
        const int r = tm * 16 + j + 8 * half;
        float v = c[j] + bv;
        if (RELU) v = fmaxf(v, 0.0f);
        out[(size_t)r * N + col] = v;  // padded rows land in d_ws, harmless
    }
}

// GATE_IDX: (64,1),(32,3),(16,5) irreps -> 240 cols map to 112 gate channels.
__device__ __forceinline__ int gate_col(int d) {
    return d < 64 ? d
         : (d < 160 ? 64 + (d - 64) / 3
                    : 96 + (d - 160) / 5);
}

// -------------------------------------------------------------------------
// out[m, e, d] = x[m, d] * gates[e, GATE_IDX[d]]
// One block = one m-row x 20 experts. x row + gate chunk staged in LDS with
// b128 loads; output streamed with 128-bit non-temporal stores (393 MB,
// ~2x L2 capacity, write-once).
// -------------------------------------------------------------------------
__global__ void broadcast_mul(const float* __restrict__ x,
                              const float* __restrict__ gates, // 112x112 (padded)
                              float* __restrict__ out)
{
    __shared__ f32x4 xs[60];            // 240 floats of x[m, :]
    __shared__ float gs[20 * 112];      // 20 expert rows of raw gates

    const int m   = blockIdx.x;
    const int e0  = blockIdx.y * 20;
    const int tid = threadIdx.x;        // blockDim = 256 (8 waves)

    if (tid < 60) xs[tid] = ((const f32x4*)(x + (size_t)m * 240))[tid];
    // 20*112 floats are one contiguous, 16B-aligned span of the gates buffer
    const f32x4* gsrc = (const f32x4*)(gates + (size_t)e0 * 112);
    for (int i = tid; i < (20 * 112) / 4; i += 256)
        ((f32x4*)gs)[i] = gsrc[i];
    __syncthreads();

    // 20 experts * 60 float4 columns = 1200 vector stores per block
    for (int idx = tid; idx < 1200; idx += 256) {
        const int el = idx / 60;
        const int q  = idx - el * 60;
        const int d0 = q * 4;
        const f32x4 xv = xs[q];
        const float* g = gs + el * 112;
        f32x4 o;
        o.x = xv.x * g[gate_col(d0 + 0)];
        o.y = xv.y * g[gate_col(d0 + 1)];
        o.z = xv.z * g[gate_col(d0 + 2)];
        o.w = xv.w * g[gate_col(d0 + 3)];
        float* dst = out + ((size_t)(m * 100 + e0 + el)) * 240 + d0;
        __builtin_nontemporal_store(o, (f32x4*)dst);
    }
}

extern "C" void kernel_launch(void* const* d_in, const int* in_sizes, int n_in,
                              void* d_out, int out_size, void* d_ws, size_t ws_size,
                              hipStream_t stream)
{
    const float* x  = (const float*)d_in[0];  // 4096 x 240
    const float* e  = (const float*)d_in[1];  // 100 x 64
    const float* W1 = (const float*)d_in[2];  // 64 x 256
    const float* b1 = (const float*)d_in[3];  // 256
    const float* W2 = (const float*)d_in[4];  // 256 x 256
    const float* b2 = (const float*)d_in[5];  // 256
    const float* W3 = (const float*)d_in[6];  // 256 x 112
    const float* b3 = (const float*)d_in[7];  // 112

    float* out = (float*)d_out;               // 4096 x 100 x 240
    float* ws  = (float*)d_ws;
    float* h1  = ws;                          // 112 x 256
    float* h2  = ws + 112 * 256;              // 112 x 256
    float* g   = ws + 2 * 112 * 256;          // 112 x 112

    // MLP (tiny; WMMA fp32). Mpad=112 -> 7 row-tiles.
    gemm_wmma_f32<true ><<<dim3(7 * 16), dim3(32), 0, stream>>>(e,  W1, b1, h1, 100, 64,  256, 16);
    gemm_wmma_f32<true ><<<dim3(7 * 16), dim3(32), 0, stream>>>(h1, W2, b2, h2, 112, 256, 256, 16);
    gemm_wmma_f32<false><<<dim3(7 * 7 ), dim3(32), 0, stream>>>(h2, W3, b3, g,  112, 256, 112, 7);

    // Store-bandwidth-bound broadcast: 4096 m-rows x 5 expert-chunks.
    broadcast_mul<<<dim3(4096, 5), dim3(256), 0, stream>>>(x, g, out);
}